// KANLayer_16758962389197
// MI455X (gfx1250) — compile-verified
//
#include <hip/hip_runtime.h>

typedef float v2f __attribute__((ext_vector_type(2)));
typedef float v8f __attribute__((ext_vector_type(8)));

#define IN_FEAT   4096
#define OUT_FEAT  2048
#define NBASIS    6
#define WPJ       8          // padded per-j weight vector: 6 basis + silu + 0
#define ITILES    16         // 2048 outputs / 128 outputs per block
#define JCHUNKS   64
#define JPC       (IN_FEAT / JCHUNKS)   // 64 j per chunk

// ---------------- Kernel 1: per-j basis + silu precompute, zero output ---------
__global__ void __launch_bounds__(256) kan_prep(const float* __restrict__ x,
                                                float* __restrict__ w,
                                                float* __restrict__ out) {
    int j = blockIdx.x * blockDim.x + threadIdx.x;
    if (j < OUT_FEAT) out[j] = 0.0f;
    if (j >= IN_FEAT) return;

    float xv = x[j];
    float t[10];
#pragma unroll
    for (int i = 0; i < 10; ++i) t[i] = -1.0f + (2.0f / 9.0f) * (float)i;

    float N[9];
#pragma unroll
    for (int i = 0; i < 9; ++i) N[i] = (xv >= t[i] && xv < t[i + 1]) ? 1.0f : 0.0f;

#pragma unroll
    for (int d = 1; d <= 3; ++d) {
#pragma unroll
        for (int i = 0; i < 9 - d; ++i) {
            float ld = t[i + d] - t[i];
            float rd = t[i + d + 1] - t[i + 1];
            float left  = (ld > 0.0f) ? (xv - t[i]) / ld * N[i] : 0.0f;
            float right = (rd > 0.0f) ? (t[i + d + 1] - xv) / rd * N[i + 1] : 0.0f;
            N[i] = left + right;
        }
    }

    float* wj = w + (size_t)j * WPJ;
#pragma unroll
    for (int k = 0; k < NBASIS; ++k) wj[k] = N[k];
    wj[6] = xv / (1.0f + expf(-xv));   // silu
    wj[7] = 0.0f;
}

// ---------------- Kernel 2: streaming WMMA reduction ---------------------------
// Data rides the A operand (per-lane-half direct loads, no selects on the load
// path); wave-uniform weights ride B (SGPR cndmask broadcast). Outputs live on
// the M axis; D is replicated across N.
//
// Layout assumptions (CDNA5 ISA 7.12.2):
//   A 16x4 f32 : lanes 0-15 -> M=lane,  {K0,K1} in VGPR{0,1}
//                lanes 16-31 -> M=lane-16, {K2,K3} in VGPR{0,1}
//   B 4x16 f32 : VGPR0 -> row K0 (lanes 0-15) / K2 (lanes 16-31); VGPR1 -> K1/K3
//   C/D        : VGPR p: lanes 0-15 = D[p, lane]; lanes 16-31 = D[p+8, lane-16]
__global__ void __launch_bounds__(256) kan_main(const float* __restrict__ coeffs,
                                                const float* __restrict__ bw,
                                                const float* __restrict__ w,
                                                float* __restrict__ out) {
    const int wave = threadIdx.x >> 5;
    const int lane = threadIdx.x & 31;
    const int m    = lane & 15;
    const bool hi  = lane >= 16;

    const int it    = blockIdx.x / JCHUNKS;
    const int jc    = blockIdx.x % JCHUNKS;
    const int ibase = it * 128 + wave * 16;
    const int i     = ibase + m;              // output index carried by this lane
    const int j0    = jc * JPC;

    // Per-lane-half direct addresses into the coeff row (24 B per (j,i)):
    //   lo lanes: {c0,c1} and c4 ; hi lanes: {c2,c3} and c5.
    const float* __restrict__ pA0 =
        coeffs + ((size_t)j0 * OUT_FEAT + i) * NBASIS + (hi ? 2 : 0);
    const float* __restrict__ pA1 =
        coeffs + ((size_t)j0 * OUT_FEAT + i) * NBASIS + 4 + (hi ? 1 : 0);
    const float* __restrict__ pBW = bw + (size_t)j0 * OUT_FEAT + i;
    const float* __restrict__ pW  = w + (size_t)j0 * WPJ;

    v8f acc = {};

    for (int jj = 0; jj < JPC; ++jj) {
        v2f   a0  = *(const v2f*)(pA0);      // lo:(c0,c1)  hi:(c2,c3), 8B aligned
        float a1x = *pA1;                     // lo: c4      hi: c5
        float bv  = *pBW;                     // base weight (hi lanes discard)
        float4 wl = *(const float4*)(pW);     // w0..w3  (wave-uniform -> SMEM)
        float4 wh = *(const float4*)(pW + 4); // w4,w5,silu,0

        // WMMA #0: K0..K3 = c0..c3 * w0..w3
        v2f b0;
        b0.x = hi ? wl.z : wl.x;              // K0=w0 (lo) / K2=w2 (hi)
        b0.y = hi ? wl.w : wl.y;              // K1=w1 (lo) / K3=w3 (hi)
        acc = __builtin_amdgcn_wmma_f32_16x16x4_f32(false, a0, false, b0,
                                                    (short)0, acc, false, false);

        // WMMA #1: K0=c4*w4, K1=bw*silu, K2=c5*w5, K3=0*0
        v2f a1, b1;
        a1.x = a1x;
        a1.y = hi ? 0.0f : bv;
        b1.x = hi ? wh.y : wh.x;              // K0=w4 (lo) / K2=w5 (hi)
        b1.y = hi ? 0.0f : wh.z;              // K1=silu (lo) / K3=0 (hi)
        acc = __builtin_amdgcn_wmma_f32_16x16x4_f32(false, a1, false, b1,
                                                    (short)0, acc, false, false);

        pA0 += (size_t)OUT_FEAT * NBASIS;
        pA1 += (size_t)OUT_FEAT * NBASIS;
        pBW += OUT_FEAT;
        pW  += WPJ;
    }

    // D replicated across N: lane 0 holds rows 0..7, lane 16 holds rows 8..15.
    if (m == 0) {
        const int base = ibase + (hi ? 8 : 0);
#pragma unroll
        for (int r = 0; r < 8; ++r) atomicAdd(&out[base + r], acc[r]);
    }
}

// ---------------- Launch --------------------------------------------------------
extern "C" void kernel_launch(void* const* d_in, const int* in_sizes, int n_in,
                              void* d_out, int out_size, void* d_ws, size_t ws_size,
                              hipStream_t stream) {
    const float* x      = (const float*)d_in[0];   // [4096]
    const float* coeffs = (const float*)d_in[1];   // [4096, 2048, 6]
    const float* bwt    = (const float*)d_in[2];   // [4096, 2048]
    float* out = (float*)d_out;                    // [2048]
    float* w   = (float*)d_ws;                     // [4096, 8] = 128 KB

    kan_prep<<<IN_FEAT / 256, 256, 0, stream>>>(x, w, out);
    kan_main<<<ITILES * JCHUNKS, 256, 0, stream>>>(coeffs, bwt, w, out);
}